// MRConv_60610578481752
// MI455X (gfx1250) — compile-verified
//
#include <hip/hip_runtime.h>

// MRConv: B=2, C=128, N=20000, K=16, OUT=128
#define C_      128
#define TWO_C   256
#define N_      20000
#define K_      16
#define OUT_    128
#define TILE_N  16
#define HSTRIDE 260   // 256 + 4 pad -> conflict-free wave32 b64 A-fragment reads
#define YSTRIDE 132   // 128 + 4 pad -> conflict-free D staging / transposed read

typedef __attribute__((ext_vector_type(2))) float v2f;
typedef __attribute__((ext_vector_type(8))) float v8f;

__device__ __forceinline__ float4 f4max(float4 a, float4 b) {
    float4 r;
    r.x = fmaxf(a.x, b.x); r.y = fmaxf(a.y, b.y);
    r.z = fmaxf(a.z, b.z); r.w = fmaxf(a.w, b.w);
    return r;
}

// x[b][c][n] -> xT[b][n][c], 32x32 LDS tiles, 256 threads
__global__ __launch_bounds__(256) void mrconv_transpose(const float* __restrict__ x,
                                                        float* __restrict__ xT) {
    __shared__ float tile[32][33];
    const int b  = blockIdx.z;
    const int n0 = blockIdx.x * 32;
    const int c0 = blockIdx.y * 32;
    const float* xp  = x  + (size_t)b * C_ * N_;
    float*       xtp = xT + (size_t)b * N_ * C_;
    const int tx = threadIdx.x & 31;
    const int ty = threadIdx.x >> 5;  // 0..7
#pragma unroll
    for (int i = 0; i < 32; i += 8)
        tile[ty + i][tx] = xp[(size_t)(c0 + ty + i) * N_ + (n0 + tx)];
    __syncthreads();
#pragma unroll
    for (int i = 0; i < 32; i += 8)
        xtp[(size_t)(n0 + ty + i) * C_ + (c0 + tx)] = tile[tx][ty + i];
}

// One block = one tile of 16 nodes for one batch. 256 threads = 8 wave32.
// Wave w computes the 16x16 output tile (16 nodes x out-channels [16w,16w+16)).
__global__ __launch_bounds__(256, 2) void mrconv_gemm(const float* __restrict__ xT,
                                                      const long long* __restrict__ edge,
                                                      const float* __restrict__ W,
                                                      const float* __restrict__ bias,
                                                      float* __restrict__ out) {
    __shared__ float sH[TILE_N * HSTRIDE];   // h tile [16][256] (reused as sY [16][YSTRIDE])
    __shared__ int   sIdx[TILE_N * K_];      // 16 nodes x 16 neighbor indices

    const int t        = threadIdx.x;
    const int tileIdx  = blockIdx.x;
    const int b        = blockIdx.y;
    const int nodeBase = tileIdx * TILE_N;
    const float* xTb   = xT + (size_t)b * N_ * C_;

    // ---- 1) neighbor indices: 256 threads == 16 nodes x 16 neighbors ----
    {
        const int node = t >> 4, k = t & 15;
        sIdx[t] = (int)edge[((size_t)b * N_ + (nodeBase + node)) * K_ + k];
    }
    __syncthreads();

    // ---- 2) gather + max-relative: h[node][c]=x_i, h[node][128+c]=max_k x_j - x_i ----
    {
        const int node = t >> 4;
        const int c0   = (t & 15) * 8;  // 8 channels per thread (2x float4)
        const float* xr = xTb + (size_t)(nodeBase + node) * C_ + c0;
        const float4 xi0 = *(const float4*)(xr);
        const float4 xi1 = *(const float4*)(xr + 4);
        const float ninf = -__builtin_inff();
        float4 m0 = make_float4(ninf, ninf, ninf, ninf);
        float4 m1 = m0;
#pragma unroll
        for (int k = 0; k < K_; ++k) {
            const int idx = sIdx[node * K_ + k];
            const float* xj = xTb + (size_t)idx * C_ + c0;
            m0 = f4max(m0, *(const float4*)(xj));
            m1 = f4max(m1, *(const float4*)(xj + 4));
        }
        float* hr = sH + node * HSTRIDE;
        *(float4*)(hr + c0)     = xi0;
        *(float4*)(hr + c0 + 4) = xi1;
        float4 r0, r1;
        r0.x = m0.x - xi0.x; r0.y = m0.y - xi0.y; r0.z = m0.z - xi0.z; r0.w = m0.w - xi0.w;
        r1.x = m1.x - xi1.x; r1.y = m1.y - xi1.y; r1.z = m1.z - xi1.z; r1.w = m1.w - xi1.w;
        *(float4*)(hr + C_ + c0)     = r0;
        *(float4*)(hr + C_ + c0 + 4) = r1;
    }
    __syncthreads();

    // ---- 3) WMMA f32 16x16x4: y[16 nodes][16 out] = h[16][256] * W[out][256]^T ----
    const int lane  = t & 31;
    const int wave  = t >> 5;
    const int oBase = wave * 16;
    const int m     = lane & 15;            // M row (A) / N col (B) within fragment
    const int koff  = (lane >> 4) << 1;     // 0 for lanes 0-15, 2 for lanes 16-31
    const float* aPtr = sH + m * HSTRIDE + koff;                   // A: h[m][k..k+1]
    const float* bPtr = W + (size_t)(oBase + m) * TWO_C + koff;    // B: W[o][k..k+1]
    v8f acc = {};
#pragma unroll 16
    for (int ks = 0; ks < 64; ++ks) {
        const int kb = ks * 4;
        const v2f a  = *(const v2f*)(aPtr + kb);   // 8B-aligned LDS pair -> ds_load_b64
        const v2f bb = *(const v2f*)(bPtr + kb);   // 8B-aligned global pair (L0/L2 hit)
        acc = __builtin_amdgcn_wmma_f32_16x16x4_f32(
            /*neg_a=*/false, a, /*neg_b=*/false, bb,
            /*c_mod=*/(short)0, acc, /*reuse_a=*/false, /*reuse_b=*/false);
    }
    __syncthreads();  // everyone done reading sH before it is reused as sY

    // ---- 4) stage D fragments to LDS: sY[node][o] ----
    float* sY = sH;
#pragma unroll
    for (int r = 0; r < 8; ++r) {
        const int node = r + ((lane >> 4) << 3);   // VGPR r: M = r (+8 for upper lanes)
        sY[node * YSTRIDE + oBase + m] = acc[r];
    }
    __syncthreads();

    // ---- 5) bias + ReLU + coalesced store: out[b][o][nodeBase+node] ----
    {
        const int o  = t >> 1;
        const int nb = (t & 1) * 8;
        const float bv = bias[o];
        float4 y0, y1;
        y0.x = fmaxf(sY[(nb + 0) * YSTRIDE + o] + bv, 0.0f);
        y0.y = fmaxf(sY[(nb + 1) * YSTRIDE + o] + bv, 0.0f);
        y0.z = fmaxf(sY[(nb + 2) * YSTRIDE + o] + bv, 0.0f);
        y0.w = fmaxf(sY[(nb + 3) * YSTRIDE + o] + bv, 0.0f);
        y1.x = fmaxf(sY[(nb + 4) * YSTRIDE + o] + bv, 0.0f);
        y1.y = fmaxf(sY[(nb + 5) * YSTRIDE + o] + bv, 0.0f);
        y1.z = fmaxf(sY[(nb + 6) * YSTRIDE + o] + bv, 0.0f);
        y1.w = fmaxf(sY[(nb + 7) * YSTRIDE + o] + bv, 0.0f);
        float* op = out + ((size_t)b * OUT_ + o) * N_ + nodeBase + nb;
        *(float4*)(op)     = y0;
        *(float4*)(op + 4) = y1;
    }
}

extern "C" void kernel_launch(void* const* d_in, const int* in_sizes, int n_in,
                              void* d_out, int out_size, void* d_ws, size_t ws_size,
                              hipStream_t stream) {
    const float*     x    = (const float*)d_in[0];      // [B, C, N, 1] f32
    const long long* edge = (const long long*)d_in[1];  // [B, N, K] i64
    const float*     W    = (const float*)d_in[2];      // [OUT, 2C] f32
    const float*     bias = (const float*)d_in[3];      // [OUT] f32
    float* out = (float*)d_out;                         // [B, OUT, N, 1] f32
    float* xT  = (float*)d_ws;                          // [B, N, C] f32 (20.48 MB)

    dim3 tgrid(N_ / 32, C_ / 32, 2);
    mrconv_transpose<<<tgrid, 256, 0, stream>>>(x, xT);

    dim3 mgrid(N_ / TILE_N, 2);
    mrconv_gemm<<<mgrid, 256, 0, stream>>>(xT, edge, W, bias, out);
}